// GraphConv_89790586290369
// MI455X (gfx1250) — compile-verified
//
#include <hip/hip_runtime.h>

typedef __attribute__((ext_vector_type(2))) float v2f;
typedef __attribute__((ext_vector_type(8))) float v8f;

#define D 128
#define LN_EPS 1e-5f

// Pair-interleaved LDS layout for B operands:
//   Wl[p*PSTRIDE + col*2 + (k&1)] holds W[k][col], p = k/2.
// One B fragment (rows k0+2*hi, k0+2*hi+1 at col) = one aligned ds_load_b64.
// PSTRIDE = 256 + 32 dwords: the +32 pad puts the two 16-lane half-groups
// (pair-rows p and p+1) on disjoint halves of the 64 LDS banks.
#define PSTRIDE 288
#define WL_SIZE (64 * PSTRIDE)   // 73728 floats? no: 64 pairs * 288 dwords

// Branchless exact-form GELU: 0.5*x*(1+erf(x/sqrt(2))).
// erf via Abramowitz-Stegun 7.1.26 (abs err <= 1.5e-7 ~ f32 roundoff).
// Uses hw v_rcp_f32 / v_exp_f32; straight-line, no EXEC divergence.
__device__ __forceinline__ float gelu_exact(float x) {
    float z  = x * 0.70710678118654752440f;
    float az = __builtin_fabsf(z);
    float t  = __builtin_amdgcn_rcpf(__builtin_fmaf(0.3275911f, az, 1.0f));
    float poly = t * (0.254829592f +
                 t * (-0.284496736f +
                 t * (1.421413741f +
                 t * (-1.453152027f +
                 t * 1.061405429f))));
    float e   = __expf(-z * z);
    float erf = __builtin_copysignf(1.0f - poly * e, z);
    return 0.5f * x * (1.0f + erf);
}

// Cooperative stage of a 128x128 f32 weight block into pair-interleaved LDS.
__device__ __forceinline__ void stage_weights(float* __restrict__ Wl,
                                              const float* __restrict__ W) {
    for (int i = threadIdx.x; i < (D * D) / 4; i += blockDim.x) {
        int k  = i >> 5;              // row (32 float4 per row)
        int c4 = (i & 31) * 4;        // starting col
        float4 w = ((const float4*)W)[i];
        float* dst = &Wl[(k >> 1) * PSTRIDE + c4 * 2 + (k & 1)];
        dst[0] = w.x; dst[2] = w.y; dst[4] = w.z; dst[6] = w.w;
    }
}

// ---------------------------------------------------------------------------
// Zero workspace (aggregated + counts)
// ---------------------------------------------------------------------------
__global__ void zero_kernel(float* __restrict__ p, long long n) {
    long long i = (long long)blockIdx.x * blockDim.x + threadIdx.x;
    if (i < n) p[i] = 0.0f;
}

// ---------------------------------------------------------------------------
// Edge kernel: edge_h = GELU(edge_f @ W_edge + b_edge)
//              messages = node_f[src] * edge_h
//              atomicAdd into aggregated[tgt], counts[tgt]
// One wave32 handles 16 edges (16x128 strip) via WMMA f32 16x16x4.
// ---------------------------------------------------------------------------
__global__ __launch_bounds__(256) void edge_kernel(
    const float* __restrict__ node_f, const long long* __restrict__ eidx,
    const float* __restrict__ edge_f, const float* __restrict__ W_edge,
    const float* __restrict__ b_edge, float* __restrict__ agg,
    float* __restrict__ counts, int E_)
{
    __shared__ float Wl[WL_SIZE];   // 72 KB pair-interleaved
    stage_weights(Wl, W_edge);
    __syncthreads();

    const int lane = threadIdx.x & 31;
    const int wave = threadIdx.x >> 5;
    const int lo   = lane & 15;         // row (A) / col-within-tile (B,C)
    const int hi   = lane >> 4;         // half-wave selector
    const long long ebase = ((long long)blockIdx.x * 8 + wave) * 16;
    if (ebase + 16 > (long long)E_) return;   // wave-uniform; EXEC stays all-1s

    v8f acc[8] = {};                    // 16x128 f32 accumulators (8 n-tiles)

    // A fragment: lane lo = row, element v holds K = k0 + 2*hi + v
    const float* arow = edge_f + (ebase + lo) * D + 2 * hi;

    #pragma unroll 2
    for (int k0 = 0; k0 < D; k0 += 4) {
        v2f a;
        a.x = arow[k0];
        a.y = arow[k0 + 1];
        // B fragment: pair-row p = k0/2 + hi, col = 16t + lo -> one b64 load
        const v2f* wb = (const v2f*)&Wl[((k0 >> 1) + hi) * PSTRIDE + lo * 2];
        #pragma unroll
        for (int t = 0; t < 8; ++t) {
            v2f b = wb[16 * t];
            acc[t] = __builtin_amdgcn_wmma_f32_16x16x4_f32(
                false, a, false, b, (short)0, acc[t], false, false);
        }
    }

    // Epilogue: bias + GELU + gather-multiply + scatter-add.
    float bias[8];
    #pragma unroll
    for (int t = 0; t < 8; ++t) bias[t] = b_edge[16 * t + lo];

    int srcR[8], tgtR[8];
    #pragma unroll
    for (int r = 0; r < 8; ++r) {
        long long e = ebase + r + 8 * hi;    // C/D layout: row = r + 8*hi
        srcR[r] = (int)eidx[e];
        tgtR[r] = (int)eidx[(long long)E_ + e];
    }

    #pragma unroll
    for (int r = 0; r < 8; ++r) {
        const float* nfr = node_f + (long long)srcR[r] * D;
        float*       agr = agg    + (long long)tgtR[r] * D;
        #pragma unroll
        for (int t = 0; t < 8; ++t) {
            int col = 16 * t + lo;
            float h = gelu_exact(acc[t][r] + bias[t]);
            atomicAdd(&agr[col], h * nfr[col]);
        }
    }
    if (lane < 16) {
        long long e = ebase + lane;
        atomicAdd(&counts[(int)eidx[(long long)E_ + e]], 1.0f);
    }
}

// ---------------------------------------------------------------------------
// Node kernel: combined = [node_f, agg/cnt]  (K = 256, two 128-wide phases)
//              out = GELU(LN(combined @ W_node + b_node))
// ---------------------------------------------------------------------------
__global__ __launch_bounds__(256) void node_kernel(
    const float* __restrict__ node_f, const float* __restrict__ agg,
    const float* __restrict__ counts, const float* __restrict__ W_node,
    const float* __restrict__ b_node, const float* __restrict__ gamma,
    const float* __restrict__ beta, float* __restrict__ out, int N_)
{
    __shared__ float Wl[WL_SIZE];   // 72 KB, reloaded per phase
    const int lane = threadIdx.x & 31;
    const int wave = threadIdx.x >> 5;
    const int lo   = lane & 15;
    const int hi   = lane >> 4;
    const long long nbase = ((long long)blockIdx.x * 8 + wave) * 16;
    const bool active = (nbase + 16 <= (long long)N_);   // wave-uniform

    v8f acc[8] = {};

    // A-fragment rows carry node index nbase+lo -> per-lane mean scale
    float scaleA = 1.0f;
    if (active) scaleA = 1.0f / fmaxf(counts[nbase + lo], 1.0f);

    for (int phase = 0; phase < 2; ++phase) {
        __syncthreads();
        stage_weights(Wl, W_node + phase * D * D);
        __syncthreads();
        if (!active) continue;   // wave-uniform branch

        const float* arow = (phase == 0 ? node_f : agg) + (nbase + lo) * D + 2 * hi;
        const float amul  = (phase == 0) ? 1.0f : scaleA;

        #pragma unroll 2
        for (int k0 = 0; k0 < D; k0 += 4) {
            v2f a;
            a.x = arow[k0] * amul;
            a.y = arow[k0 + 1] * amul;
            const v2f* wb = (const v2f*)&Wl[((k0 >> 1) + hi) * PSTRIDE + lo * 2];
            #pragma unroll
            for (int t = 0; t < 8; ++t) {
                v2f b = wb[16 * t];
                acc[t] = __builtin_amdgcn_wmma_f32_16x16x4_f32(
                    false, a, false, b, (short)0, acc[t], false, false);
            }
        }
    }
    if (!active) return;

    float bn[8], g[8], be[8];
    #pragma unroll
    for (int t = 0; t < 8; ++t) {
        int col = 16 * t + lo;
        bn[t] = b_node[col];
        g[t]  = gamma[col];
        be[t] = beta[col];
    }
    #pragma unroll
    for (int t = 0; t < 8; ++t)
        #pragma unroll
        for (int r = 0; r < 8; ++r)
            acc[t][r] += bn[t];

    // Row-wise LayerNorm: row m = r + 8*hi lives in a 16-lane half-group
    // (8 accumulator regs x 16 lanes). Butterfly reduce within 16 lanes.
    #pragma unroll
    for (int r = 0; r < 8; ++r) {
        float s = 0.0f;
        #pragma unroll
        for (int t = 0; t < 8; ++t) s += acc[t][r];
        #pragma unroll
        for (int off = 1; off < 16; off <<= 1) s += __shfl_xor(s, off, 32);
        float mean = s * (1.0f / D);

        float q = 0.0f;
        #pragma unroll
        for (int t = 0; t < 8; ++t) {
            float d = acc[t][r] - mean;
            q += d * d;
        }
        #pragma unroll
        for (int off = 1; off < 16; off <<= 1) q += __shfl_xor(q, off, 32);
        float rstd = rsqrtf(q * (1.0f / D) + LN_EPS);

        float* orow = out + (nbase + r + 8 * hi) * D;
        #pragma unroll
        for (int t = 0; t < 8; ++t) {
            float y = (acc[t][r] - mean) * rstd * g[t] + be[t];
            orow[16 * t + lo] = gelu_exact(y);
        }
    }
}

// ---------------------------------------------------------------------------
extern "C" void kernel_launch(void* const* d_in, const int* in_sizes, int n_in,
                              void* d_out, int out_size, void* d_ws, size_t ws_size,
                              hipStream_t stream) {
    const float*     node_f = (const float*)d_in[0];
    const long long* eidx   = (const long long*)d_in[1];   // int64 (2, E)
    const float*     edge_f = (const float*)d_in[2];
    const float*     W_edge = (const float*)d_in[3];
    const float*     b_edge = (const float*)d_in[4];
    const float*     W_node = (const float*)d_in[5];
    const float*     b_node = (const float*)d_in[6];
    const float*     gamma  = (const float*)d_in[7];
    const float*     beta   = (const float*)d_in[8];

    const int N = in_sizes[0] / D;
    const int E = in_sizes[2] / D;

    float* agg    = (float*)d_ws;              // N*D
    float* counts = agg + (size_t)N * D;       // N

    long long zn = (long long)N * (D + 1);
    zero_kernel<<<(int)((zn + 255) / 256), 256, 0, stream>>>(agg, zn);

    int ewaves  = (E + 15) / 16;
    int eblocks = (ewaves + 7) / 8;
    edge_kernel<<<eblocks, 256, 0, stream>>>(node_f, eidx, edge_f, W_edge,
                                             b_edge, agg, counts, E);

    int nwaves  = (N + 15) / 16;
    int nblocks = (nwaves + 7) / 8;
    node_kernel<<<nblocks, 256, 0, stream>>>(node_f, agg, counts, W_node,
                                             b_node, gamma, beta,
                                             (float*)d_out, N);
}